// Softmax_81570018885750
// MI455X (gfx1250) — compile-verified
//
#include <hip/hip_runtime.h>
#include <stdint.h>

// Masked/biased/scaled softmax, [B,H,Q,K] = [2,16,2048,2048] f32.
// Bandwidth-bound: min traffic 1.31 GB -> ~56us @ 23.3 TB/s.
// Block = (q, both b, 4 h) = 8 rows; bias read once (reused over b in regs),
// mask read once per thread (reused over h in regs), input streamed via
// CDNA5 async global->LDS b128 copies (ASYNCcnt), values staged in big LDS.

#ifndef __has_builtin
#define __has_builtin(x) 0
#endif

#define AS1 __attribute__((address_space(1)))
#define AS3 __attribute__((address_space(3)))

typedef int v4i __attribute__((ext_vector_type(4)));
typedef float v4f __attribute__((ext_vector_type(4)));

namespace {

constexpr int B_ = 2, H_ = 16, Q_ = 2048, K_ = 2048;
constexpr int THREADS = 256;
constexpr float SCALE = 0.08838834764831845f;
constexpr float MASK_NEG = -1e10f;

template <int N>
__device__ __forceinline__ void wait_asynccnt() {
  asm volatile("s_wait_asynccnt %0" ::"i"(N) : "memory");
}

// Async copy of 16 bytes per lane, global -> LDS (tracked by ASYNCcnt).
__device__ __forceinline__ void async_copy16(const float* g, float* l) {
#if __has_builtin(__builtin_amdgcn_global_load_async_to_lds_b128)
  __builtin_amdgcn_global_load_async_to_lds_b128(
      (AS1 v4i*)(uintptr_t)g, (AS3 v4i*)(uint32_t)(uintptr_t)l, 0, 0);
#else
  uint32_t laddr = (uint32_t)(uintptr_t)l;
  asm volatile("global_load_async_to_lds_b128 %0, %1, off" ::"v"(laddr), "v"(g)
               : "memory");
#endif
}

__device__ __forceinline__ float wave_max(float v) {
#pragma unroll
  for (int off = 16; off > 0; off >>= 1) v = fmaxf(v, __shfl_xor(v, off, 32));
  return v;
}

__device__ __forceinline__ float wave_sum(float v) {
#pragma unroll
  for (int off = 16; off > 0; off >>= 1) v += __shfl_xor(v, off, 32);
  return v;
}

__global__ void __launch_bounds__(THREADS)
masked_softmax_kernel(const float* __restrict__ in, const int* __restrict__ mask,
                      const float* __restrict__ bias, float* __restrict__ out) {
  __shared__ __align__(16) float vals[8][K_];      // 64 KB: masked values
  __shared__ __align__(16) float stage[2][2][K_];  // 32 KB: async input stage
  __shared__ float red[8][THREADS / 32];           // wave partials
  __shared__ float rowstat[8];                     // row max, then 1/sum

  const int t = threadIdx.x;
  const int lane = t & 31;
  const int wave = t >> 5;
  const int q = blockIdx.y;
  const int h0 = blockIdx.x * 4;
  const int k0 = t * 4;        // first float4 chunk
  const int k1 = k0 + K_ / 2;  // second float4 chunk

  // Mask loaded once, reused for all 4 h iterations (broadcast over H).
  const v4i m0a = *(const v4i*)(mask + ((size_t)0 * Q_ + q) * K_ + k0);
  const v4i m0b = *(const v4i*)(mask + ((size_t)0 * Q_ + q) * K_ + k1);
  const v4i m1a = *(const v4i*)(mask + ((size_t)1 * Q_ + q) * K_ + k0);
  const v4i m1b = *(const v4i*)(mask + ((size_t)1 * Q_ + q) * K_ + k1);

  auto issue = [&](int i, int buf) {  // prefetch input rows (b=0,1) for h0+i
    const int h = h0 + i;
    const size_t r0 = (((size_t)(0 * H_ + h)) * Q_ + q) * (size_t)K_;
    const size_t r1 = (((size_t)(1 * H_ + h)) * Q_ + q) * (size_t)K_;
    async_copy16(in + r0 + k0, &stage[buf][0][k0]);
    async_copy16(in + r0 + k1, &stage[buf][0][k1]);
    async_copy16(in + r1 + k0, &stage[buf][1][k0]);
    async_copy16(in + r1 + k1, &stage[buf][1][k1]);
  };

  auto msel4 = [](v4i m, v4f x, v4f b) {
    v4f v;
    v.x = m.x ? MASK_NEG : (x.x + b.x) * SCALE;
    v.y = m.y ? MASK_NEG : (x.y + b.y) * SCALE;
    v.z = m.z ? MASK_NEG : (x.z + b.z) * SCALE;
    v.w = m.w ? MASK_NEG : (x.w + b.w) * SCALE;
    return v;
  };
  auto max4 = [](v4f v) { return fmaxf(fmaxf(v.x, v.y), fmaxf(v.z, v.w)); };
  auto sum4 = [](v4f v) { return (v.x + v.y) + (v.z + v.w); };

  issue(0, 0);
  issue(1, 1);

  // Pass 1: combine input+bias+mask, stash f32 values in LDS, per-row max.
#pragma unroll
  for (int i = 0; i < 4; ++i) {
    const int buf = i & 1;
    const int h = h0 + i;
    if (i < 3) wait_asynccnt<4>();  // this iter's batch done (next in flight)
    else       wait_asynccnt<0>();

    const v4f ba = *(const v4f*)(bias + ((size_t)h * Q_ + q) * K_ + k0);
    const v4f bb = *(const v4f*)(bias + ((size_t)h * Q_ + q) * K_ + k1);

    {  // b = 0 -> row i
      v4f va = msel4(m0a, *(const v4f*)&stage[buf][0][k0], ba);
      v4f vb = msel4(m0b, *(const v4f*)&stage[buf][0][k1], bb);
      *(v4f*)&vals[i][k0] = va;
      *(v4f*)&vals[i][k1] = vb;
      float m = wave_max(fmaxf(max4(va), max4(vb)));
      if (lane == 0) red[i][wave] = m;
    }
    {  // b = 1 -> row 4+i (same bias registers: reuse over b)
      v4f va = msel4(m1a, *(const v4f*)&stage[buf][1][k0], ba);
      v4f vb = msel4(m1b, *(const v4f*)&stage[buf][1][k1], bb);
      *(v4f*)&vals[4 + i][k0] = va;
      *(v4f*)&vals[4 + i][k1] = vb;
      float m = wave_max(fmaxf(max4(va), max4(vb)));
      if (lane == 0) red[4 + i][wave] = m;
    }
    if (i < 2) issue(i + 2, buf);  // refill the buffer we just drained
  }

  __syncthreads();
  if (t < 8) {
    float m = red[t][0];
#pragma unroll
    for (int w = 1; w < 8; ++w) m = fmaxf(m, red[t][w]);
    rowstat[t] = m;
  }
  __syncthreads();

  // Pass 2: exp(v - rowmax) in LDS, per-row sum.
#pragma unroll
  for (int r = 0; r < 8; ++r) {
    const float rm = rowstat[r];
    v4f a = *(const v4f*)&vals[r][k0];
    v4f c = *(const v4f*)&vals[r][k1];
    a.x = __expf(a.x - rm); a.y = __expf(a.y - rm);
    a.z = __expf(a.z - rm); a.w = __expf(a.w - rm);
    c.x = __expf(c.x - rm); c.y = __expf(c.y - rm);
    c.z = __expf(c.z - rm); c.w = __expf(c.w - rm);
    *(v4f*)&vals[r][k0] = a;
    *(v4f*)&vals[r][k1] = c;
    float s = wave_sum(sum4(a) + sum4(c));
    if (lane == 0) red[r][wave] = s;
  }
  __syncthreads();
  if (t < 8) {
    float s = 0.0f;
#pragma unroll
    for (int w = 0; w < 8; ++w) s += red[t][w];
    rowstat[t] = 1.0f / s;
  }
  __syncthreads();

  // Pass 3: normalize and stream out (non-temporal: no reuse).
#pragma unroll
  for (int r = 0; r < 8; ++r) {
    const int b = r >> 2;
    const int h = h0 + (r & 3);
    const float inv = rowstat[r];
    v4f a = *(const v4f*)&vals[r][k0];
    v4f c = *(const v4f*)&vals[r][k1];
    a *= inv;
    c *= inv;
    const size_t ob = (((size_t)(b * H_ + h)) * Q_ + q) * (size_t)K_;
    __builtin_nontemporal_store(a, (v4f*)(out + ob + k0));
    __builtin_nontemporal_store(c, (v4f*)(out + ob + k1));
  }
}

}  // namespace

extern "C" void kernel_launch(void* const* d_in, const int* in_sizes, int n_in,
                              void* d_out, int out_size, void* d_ws,
                              size_t ws_size, hipStream_t stream) {
  (void)in_sizes; (void)n_in; (void)out_size; (void)d_ws; (void)ws_size;
  const float* in = (const float*)d_in[0];
  const int* mask = (const int*)d_in[1];
  const float* bias = (const float*)d_in[2];
  float* out = (float*)d_out;
  dim3 grid(H_ / 4, Q_);  // x = h-quad (fast: mask-sharing blocks co-resident)
  masked_softmax_kernel<<<grid, THREADS, 0, stream>>>(in, mask, bias, out);
}